// SinkAttentionRotaryImpl_11759620456496
// MI455X (gfx1250) — compile-verified
//
#include <hip/hip_runtime.h>
#include <math.h>
#include <stdint.h>

// ---- problem constants (from reference) ----
#define CACHE_SIZE       (4096 + 16)   // SLIDING_WINDOW + SINK_SIZE
#define BLOCK_FLOATS     16384         // 8 heads * 16 * 16 * 8 floats per cache block
#define MAX_BLOCKS_SEQ   128
#define BATCH            64

// ---- copy kernel config ----
#define COPY_GRID    2048
#define COPY_TPB     256
#define COPY_UNROLL  4                     // float4's per thread per group
#define SLOT_BYTES   (COPY_TPB * 16)       // LDS stride between unroll slots

// -----------------------------------------------------------------------------
// Async copy helpers: CDNA5 global<->LDS DMA path (ASYNCcnt), no VGPR data.
// -----------------------------------------------------------------------------
__device__ __forceinline__ void async_load_group(const float4* __restrict__ src,
                                                 long long base, long long gsize,
                                                 long long n4, unsigned lds_grp) {
#pragma unroll
  for (int u = 0; u < COPY_UNROLL; ++u) {
    long long i = base + (long long)u * gsize;
    if (i < n4) {
      const float4* ga = src + i;
      unsigned la = lds_grp + (unsigned)(u * SLOT_BYTES);
      asm volatile("global_load_async_to_lds_b128 %0, %1, off"
                   :
                   : "v"(la), "v"(ga)
                   : "memory");
    }
  }
}

__device__ __forceinline__ void async_store_group(float4* __restrict__ dst,
                                                  long long base, long long gsize,
                                                  long long n4, unsigned lds_grp) {
#pragma unroll
  for (int u = 0; u < COPY_UNROLL; ++u) {
    long long i = base + (long long)u * gsize;
    if (i < n4) {
      float4* ga = dst + i;
      unsigned la = lds_grp + (unsigned)(u * SLOT_BYTES);
      asm volatile("global_store_async_from_lds_b128 %0, %1, off"
                   :
                   : "v"(ga), "v"(la)
                   : "memory");
    }
  }
}

// -----------------------------------------------------------------------------
// Kernel 1: bulk stream copy, double-buffered software pipeline.
// Each s_wait_asynccnt-0 barrier covers one load-group AND one store-group in
// flight concurrently (8 outstanding async ops per wave).
//   buf A: wait -> store A, load B (other LDS region) -> wait -> store B, load A ...
// Final stores are drained by S_ENDPGM's implicit wait-idle.
// -----------------------------------------------------------------------------
__global__ __launch_bounds__(COPY_TPB) void copy_async_kernel(
    const float4* __restrict__ src, float4* __restrict__ dst, long long n4) {
  __shared__ __align__(16) float4 stage[COPY_TPB * COPY_UNROLL * 2];  // 32 KB

  const int tid = threadIdx.x;
  // Low 32 bits of the flat shared-aperture address == LDS byte offset
  // (ISA 10.2).
  const unsigned lds0      = (unsigned)(uintptr_t)(void*)&stage[tid];
  const unsigned bufstride = (unsigned)(COPY_TPB * COPY_UNROLL * 16);

  const long long gsize = (long long)gridDim.x * COPY_TPB;
  const long long chunk = gsize * COPY_UNROLL;
  long long base        = (long long)blockIdx.x * COPY_TPB + tid;

  unsigned buf = 0;
  // Prologue: prime buffer 0.
  async_load_group(src, base, gsize, n4, lds0);

  for (; base < n4; base += chunk) {
    // Loads into `buf` (and stores from the other buffer issued last
    // iteration) have completed.
    asm volatile("s_wait_asynccnt 0x0" ::: "memory");

    const long long nxt = base + chunk;
    if (nxt < n4) {
      // Prefetch next group into the other LDS buffer (no overlap with the
      // region the upcoming stores read).
      async_load_group(src, nxt, gsize, n4, lds0 + (buf ^ 1u) * bufstride);
    }
    // Drain the completed buffer to the destination.
    async_store_group(dst, base, gsize, n4, lds0 + buf * bufstride);

    buf ^= 1u;
  }
}

// -----------------------------------------------------------------------------
// Kernel 2: RoPE-rotate the 64 sink blocks (reads pristine input, overwrites
// the already-copied output blocks). One workgroup per batch.
//
// Layout inside a cache block (floats): [h][d_hi][tok][d_lo] with strides
// 2048 / 128 / 8 / 1. Logical dim d = d_hi*8 + d_lo; rotation pairs d <-> d+64,
// i.e. (h,d_hi,tok,d_lo) <-> (h,d_hi+8,tok,d_lo): flat stride +1024.
// Frequency index j = d_hi*8 + d_lo, j in [0,64).
// -----------------------------------------------------------------------------
__global__ __launch_bounds__(256) void rotate_sinks_kernel(
    const float* __restrict__ key_cache, const int* __restrict__ positions,
    const int* __restrict__ block_tables, float* __restrict__ out) {
  __shared__ float s_cos[64];
  __shared__ float s_sin[64];

  const int b   = blockIdx.x;
  const int tid = threadIdx.x;

  int evict = positions[b] - CACHE_SIZE;
  if (evict < 0) evict = 0;

  const long long sink = (long long)block_tables[(long long)b * MAX_BLOCKS_SEQ];
  const long long base = sink * (long long)BLOCK_FLOATS;

  if (tid < 64) {
    // inv_freq[j] = 10000^(-j/64); correctly-rounded via double exp2, then the
    // angle is formed in f32 to mirror the reference's f32 pipeline.
    // log2(10000) = 13.287712379549449
    float invf = (float)exp2(-(double)tid * (13.287712379549449 / 64.0));
    float ang  = (float)evict * invf;
    float s, c;
    sincosf(ang, &s, &c);
    s_cos[tid] = c;
    s_sin[tid] = s;
  }
  __syncthreads();

  // 8192 rotation pairs per block: p = h*1024 + d_hi*128 + tok*8 + d_lo
  for (int p = tid; p < 8192; p += 256) {
    const int h   = p >> 10;
    const int rem = p & 1023;                      // d_hi*128 + tok*8 + d_lo
    const int j   = ((rem >> 7) << 3) | (rem & 7); // d_hi*8 + d_lo

    const long long i1 = base + (long long)h * 2048 + rem;
    const long long i2 = i1 + 1024;                // d_hi + 8

    const float x1 = key_cache[i1];
    const float x2 = key_cache[i2];
    const float c  = s_cos[j];
    const float s  = s_sin[j];

    out[i1] = x1 * c - x2 * s;
    out[i2] = x2 * c + x1 * s;
  }
}

// -----------------------------------------------------------------------------
extern "C" void kernel_launch(void* const* d_in, const int* in_sizes, int n_in,
                              void* d_out, int out_size, void* d_ws, size_t ws_size,
                              hipStream_t stream) {
  const float* key_cache    = (const float*)d_in[0];
  const int*   positions    = (const int*)d_in[1];
  const int*   block_tables = (const int*)d_in[2];
  float*       out          = (float*)d_out;

  const long long n  = (long long)out_size; // 134,217,728 floats (512 MB)
  const long long n4 = n >> 2;              // float4 count

  copy_async_kernel<<<COPY_GRID, COPY_TPB, 0, stream>>>(
      (const float4*)key_cache, (float4*)out, n4);

  rotate_sinks_kernel<<<BATCH, 256, 0, stream>>>(
      key_cache, positions, block_tables, out);
}